// LossHybrid_29927332119238
// MI455X (gfx1250) — compile-verified
//
#include <hip/hip_runtime.h>
#include <math.h>

#define BB 16
#define TOTAL 33600
#define NGT 32
#define NC 80
#define PSTR 84          // 4 box + 80 logits
#define N_CAND 50
#define EPSF 1e-7f

typedef float v2f __attribute__((ext_vector_type(2)));
typedef float v8f __attribute__((ext_vector_type(8)));

#if __has_builtin(__builtin_amdgcn_global_load_async_to_lds_b128) && \
    __has_builtin(__builtin_amdgcn_s_wait_asynccnt)
#define USE_ASYNC_LDS 1
typedef int av4i __attribute__((ext_vector_type(4)));
#else
#define USE_ASYNC_LDS 0
#endif

__device__ __forceinline__ float softplusf(float x) {
    // numerically stable log(1+exp(x))
    return fmaxf(x, 0.0f) + log1pf(expf(-fabsf(x)));
}

// ---------------------------------------------------------------------------
// k0: zero/init scratch, precompute per-anchor xyxy box and center
// ---------------------------------------------------------------------------
__global__ void k0_init_prep(const float* __restrict__ p_flat,
                             float* __restrict__ pbox, float* __restrict__ pc,
                             unsigned* __restrict__ candmask,
                             unsigned* __restrict__ validf,
                             unsigned long long* __restrict__ bestkey,
                             unsigned* __restrict__ conflict,
                             float* __restrict__ acc) {
    int i = blockIdx.x * blockDim.x + threadIdx.x;
    if (i >= BB * TOTAL) return;
    const float* p = p_flat + (size_t)i * PSTR;
    float cx = p[0], cy = p[1], w = p[2], h = p[3];
    pbox[4 * i + 0] = cx - 0.5f * w;
    pbox[4 * i + 1] = cy - 0.5f * h;
    pbox[4 * i + 2] = cx + 0.5f * w;
    pbox[4 * i + 3] = cy + 0.5f * h;
    pc[2 * i + 0] = cx;
    pc[2 * i + 1] = cy;
    candmask[i] = 0u;
    validf[i] = 0u;
    if (i < BB * NGT) bestkey[i] = ~0ull;
    if (i < BB) conflict[i] = 0u;
    if (i < 4) acc[i] = 0.0f;
}

// ---------------------------------------------------------------------------
// k1: squared center-distance matrix via fp32 WMMA, stored TRANSPOSED:
//   distT[b][gt][anchor]  (contiguous per-GT columns for the top-k scan).
//   dist(a,g) = |pc|^2 + |gc|^2 - 2 pc.gc  as a K=4 GEMM:
//   A row (16x4) = [px, py, |p|^2, 1],  B col (4x16) = [-2gx, -2gy, 1, |g|^2]
// One wave per 16-anchor tile; two N-tiles cover NGT=32.
// A layout (ISA 7.12.2, 32-bit A 16x4): lanes0-15 hold K0/K1, lanes16-31 K2/K3.
// D layout: lanes0-15 -> M=r, N=lane ; lanes16-31 -> M=8+r, N=lane-16.
// Each lane thus owns 8 consecutive anchors of one column -> two b128 stores.
// ---------------------------------------------------------------------------
__global__ void k1_dist_wmma(const float* __restrict__ pc,
                             const float* __restrict__ gt_boxes,
                             float* __restrict__ distT) {
    int tile = blockIdx.x;          // 0..2099
    int b    = blockIdx.y;          // 0..15
    int lane = threadIdx.x;         // 0..31, wave32
    int m    = lane & 15;
    int half = lane >> 4;

    int anchor = tile * 16 + m;
    const float* pcb = pc + ((size_t)b * TOTAL + anchor) * 2;
    float px = pcb[0], py = pcb[1];
    v2f a;
    if (half == 0) { a.x = px;                 a.y = py;   }  // K=0,1
    else           { a.x = px * px + py * py;  a.y = 1.0f; }  // K=2,3

    float* dT = distT + (size_t)b * NGT * TOTAL;

    #pragma unroll
    for (int nt = 0; nt < 2; ++nt) {
        int j = nt * 16 + m;
        const float* g = gt_boxes + ((size_t)b * NGT + j) * 4;
        float gx = 0.5f * (g[0] + g[2]);
        float gy = 0.5f * (g[1] + g[3]);
        v2f bm;
        if (half == 0) { bm.x = -2.0f * gx; bm.y = -2.0f * gy;       } // K=0,1
        else           { bm.x = 1.0f;       bm.y = gx * gx + gy * gy;} // K=2,3
        v8f c = {};
        c = __builtin_amdgcn_wmma_f32_16x16x4_f32(
                false, a, false, bm, (short)0, c, false, false);
        // this lane: column (nt*16 + m), anchors tile*16 + half*8 + [0..7]
        float* colp = dT + (size_t)(nt * 16 + m) * TOTAL + tile * 16 + half * 8;
        float4* v4 = (float4*)colp;                 // 32B aligned
        v4[0] = make_float4(c[0], c[1], c[2], c[3]);
        v4[1] = make_float4(c[4], c[5], c[6], c[7]);
    }
}

// ---------------------------------------------------------------------------
// k2: exact top-N_CAND per (image, GT): stage the contiguous column in LDS
// (one coalesced pass; async global->LDS when the builtin exists), then run
// 50 successive lexicographic-minimum rounds entirely out of LDS.
// ---------------------------------------------------------------------------
__global__ void k2_topk(const float* __restrict__ distT,
                        unsigned* __restrict__ candmask) {
    int j = blockIdx.x;             // GT index
    int b = blockIdx.y;             // image
    int tid = threadIdx.x;          // 0..255
    const float* col = distT + ((size_t)b * NGT + j) * TOTAL;
    unsigned* cm = candmask + (size_t)b * TOTAL;

    __shared__ float lcol[TOTAL];   // 134.4 KB (< 320 KB CDNA5 LDS)
    __shared__ float sval[256];
    __shared__ int   sidx[256];

#if USE_ASYNC_LDS
    for (int i = tid * 4; i < TOTAL; i += 256 * 4) {
        __builtin_amdgcn_global_load_async_to_lds_b128(
            (av4i*)(col + i), (av4i*)&lcol[i], 0, 0);
    }
    __builtin_amdgcn_s_wait_asynccnt(0);
    __syncthreads();
#else
    for (int i = tid; i < TOTAL; i += 256) lcol[i] = col[i];
    __syncthreads();
#endif

    float lastV = -1.0f;            // all distances >= 0
    int   lastI = -1;

    for (int k = 0; k < N_CAND; ++k) {
        float bestV = 3.0e38f;
        int   bestI = 0x7fffffff;
        for (int i = tid; i < TOTAL; i += 256) {
            float v = lcol[i];
            bool after_last = (v > lastV) || (v == lastV && i > lastI);
            if (after_last && (v < bestV || (v == bestV && i < bestI))) {
                bestV = v; bestI = i;
            }
        }
        sval[tid] = bestV; sidx[tid] = bestI;
        __syncthreads();
        for (int s = 128; s > 0; s >>= 1) {
            if (tid < s) {
                if (sval[tid + s] < sval[tid] ||
                    (sval[tid + s] == sval[tid] && sidx[tid + s] < sidx[tid])) {
                    sval[tid] = sval[tid + s]; sidx[tid] = sidx[tid + s];
                }
            }
            __syncthreads();
        }
        lastV = sval[0]; lastI = sidx[0];
        __syncthreads();
        if (tid == 0 && lastI >= 0 && lastI < TOTAL) cm[lastI] = 1u;
    }
}

// ---------------------------------------------------------------------------
// k3: for candidate anchors only: full 32-way cost row.
//   per_anchor_gt = row argmin; per-GT column argmin via packed atomicMin.
// cost > 0 so float bits are order-preserving; tie-break on lower anchor idx.
// ---------------------------------------------------------------------------
__global__ void k3_cost(const float* __restrict__ p_flat,
                        const float* __restrict__ gt_boxes,
                        const int*   __restrict__ gt_labels,
                        const float* __restrict__ pbox,
                        const unsigned* __restrict__ candmask,
                        int* __restrict__ pagt,
                        unsigned long long* __restrict__ bestkey) {
    int i = blockIdx.x * blockDim.x + threadIdx.x;
    if (i >= BB * TOTAL) return;
    if (!candmask[i]) return;
    int b = i / TOTAL;
    int a = i - b * TOTAL;

    const float* pb = pbox + (size_t)i * 4;
    float px1 = pb[0], py1 = pb[1], px2 = pb[2], py2 = pb[3];
    float a1 = fmaxf(px2 - px1, 0.0f) * fmaxf(py2 - py1, 0.0f);
    const float* logits = p_flat + (size_t)i * PSTR + 4;

    float bestC = 3.0e38f;
    int   bestJ = 0;
    for (int j = 0; j < NGT; ++j) {
        const float* g = gt_boxes + ((size_t)b * NGT + j) * 4;
        float gx1 = g[0], gy1 = g[1], gx2 = g[2], gy2 = g[3];
        float a2 = fmaxf(gx2 - gx1, 0.0f) * fmaxf(gy2 - gy1, 0.0f);
        float iw = fmaxf(fminf(px2, gx2) - fmaxf(px1, gx1), 0.0f);
        float ih = fmaxf(fminf(py2, gy2) - fmaxf(py1, gy1), 0.0f);
        float inter = iw * ih;
        float iou = inter / (a1 + a2 - inter + EPSF);
        int lj = gt_labels[b * NGT + j];
        float x = logits[lj];
        float ccls = softplusf(x) - x;              // BCE with target 1
        float cost = ccls + 6.0f * (1.0f - iou) + 1e-6f;
        if (cost < bestC) { bestC = cost; bestJ = j; }
        unsigned long long key =
            ((unsigned long long)__float_as_uint(cost) << 32) | (unsigned)a;
        atomicMin(&bestkey[b * NGT + j], key);
    }
    pagt[i] = bestJ;
}

// ---------------------------------------------------------------------------
// k4: per image: detect best-anchor conflicts; no-conflict -> valid = matches1
// ---------------------------------------------------------------------------
__global__ void k4_finalize(const unsigned long long* __restrict__ bestkey,
                            unsigned* __restrict__ conflict,
                            unsigned* __restrict__ validf) {
    int b = blockIdx.x;
    int j = threadIdx.x;            // 0..31
    __shared__ int ba[NGT];
    __shared__ int cf;
    if (j == 0) cf = 0;
    __syncthreads();
    int mya = (int)(bestkey[b * NGT + j] & 0xffffffffull);
    ba[j] = mya;
    __syncthreads();
    for (int j2 = 0; j2 < NGT; ++j2)
        if (j2 != j && ba[j2] == mya) cf = 1;   // benign race, all write 1
    __syncthreads();
    if (j == 0) conflict[b] = (unsigned)cf;
    if (!cf && mya >= 0 && mya < TOTAL)
        validf[(size_t)b * TOTAL + mya] = 1u;
}

// ---------------------------------------------------------------------------
// k5: conflict branch: valid = cand_mask  (matches2.any(axis=1) == cand_mask)
// ---------------------------------------------------------------------------
__global__ void k5_valid(const unsigned* __restrict__ conflict,
                         const unsigned* __restrict__ candmask,
                         unsigned* __restrict__ validf) {
    int i = blockIdx.x * blockDim.x + threadIdx.x;
    if (i >= BB * TOTAL) return;
    int b = i / TOTAL;
    if (conflict[b]) validf[i] = candmask[i];
}

// ---------------------------------------------------------------------------
// k6: accumulate lbox (1 - CIoU), npos, and the sparse lcls correction
// ---------------------------------------------------------------------------
__global__ void k6_loss(const float* __restrict__ p_flat,
                        const float* __restrict__ gt_boxes,
                        const int*   __restrict__ gt_labels,
                        const float* __restrict__ pbox,
                        const unsigned* __restrict__ validf,
                        const int* __restrict__ pagt,
                        float* __restrict__ acc) {
    int i = blockIdx.x * blockDim.x + threadIdx.x;
    if (i >= BB * TOTAL) return;
    if (!validf[i]) return;
    int b = i / TOTAL;
    int j = pagt[i];
    const float* pb = pbox + (size_t)i * 4;
    const float* g  = gt_boxes + ((size_t)b * NGT + j) * 4;
    float px1 = pb[0], py1 = pb[1], px2 = pb[2], py2 = pb[3];
    float tx1 = g[0],  ty1 = g[1],  tx2 = g[2],  ty2 = g[3];

    float iw = fmaxf(fminf(px2, tx2) - fmaxf(px1, tx1), 0.0f);
    float ih = fmaxf(fminf(py2, ty2) - fmaxf(py1, ty1), 0.0f);
    float inter = iw * ih;
    float w1 = px2 - px1, h1 = py2 - py1;
    float w2 = tx2 - tx1, h2 = ty2 - ty1;
    float uni = w1 * h1 + w2 * h2 - inter + EPSF;
    float iou = inter / uni;
    float cw = fmaxf(px2, tx2) - fminf(px1, tx1);
    float ch = fmaxf(py2, ty2) - fminf(py1, ty1);
    float c2 = cw * cw + ch * ch + EPSF;
    float dx = px1 + px2 - tx1 - tx2;
    float dy = py1 + py2 - ty1 - ty2;
    float rho2 = (dx * dx + dy * dy) * 0.25f;
    float dat = atanf(w2 / (h2 + EPSF)) - atanf(w1 / (h1 + EPSF));
    float v = (4.0f / (3.14159f * 3.14159f)) * dat * dat;
    float alpha = v / (1.0f - iou + v + EPSF);
    float ciou = fminf(fmaxf(iou - rho2 / c2 - alpha * v, -1.0f), 1.0f);

    atomicAdd(&acc[0], 1.0f - ciou);                       // lbox
    int lj = gt_labels[b * NGT + j];
    float x = p_flat[(size_t)i * PSTR + 4 + lj];
    atomicAdd(&acc[1], x);                                 // lcls correction
    atomicAdd(&acc[2], 1.0f);                              // npos
}

// ---------------------------------------------------------------------------
// k7: sum of softplus over all B*TOTAL*80 logits (dominant HBM traffic)
// ---------------------------------------------------------------------------
__global__ void k7_softplus(const float* __restrict__ p_flat,
                            float* __restrict__ acc) {
    int i = blockIdx.x * blockDim.x + threadIdx.x;
    float s = 0.0f;
    if (i < BB * TOTAL) {
        const float4* p = (const float4*)(p_flat + (size_t)i * PSTR + 4);
        if (i + 1 < BB * TOTAL)
            __builtin_prefetch(p_flat + (size_t)(i + 1) * PSTR + 4, 0, 0);
        #pragma unroll
        for (int q = 0; q < NC / 4; ++q) {
            float4 v = p[q];
            s += softplusf(v.x) + softplusf(v.y) +
                 softplusf(v.z) + softplusf(v.w);
        }
    }
    __shared__ float red[256];
    red[threadIdx.x] = s;
    __syncthreads();
    for (int st = 128; st > 0; st >>= 1) {
        if (threadIdx.x < st) red[threadIdx.x] += red[threadIdx.x + st];
        __syncthreads();
    }
    if (threadIdx.x == 0) atomicAdd(&acc[3], red[0]);
}

// ---------------------------------------------------------------------------
// k8: final scalar
// ---------------------------------------------------------------------------
__global__ void k8_final(const float* __restrict__ acc, float* __restrict__ out) {
    float norm = fmaxf(1.0f, acc[2]);
    out[0] = acc[0] * 7.5f / norm + (acc[3] - acc[1]) * 0.5f / norm;
}

// ---------------------------------------------------------------------------
extern "C" void kernel_launch(void* const* d_in, const int* in_sizes, int n_in,
                              void* d_out, int out_size, void* d_ws, size_t ws_size,
                              hipStream_t stream) {
    const float* p_flat    = (const float*)d_in[0];   // B*TOTAL*84
    const float* gt_boxes  = (const float*)d_in[1];   // B*NGT*4
    const int*   gt_labels = (const int*)d_in[2];     // B*NGT
    float* out = (float*)d_out;

    // Carve workspace (256B aligned slices)
    char* w = (char*)d_ws;
    auto carve = [&](size_t bytes) {
        char* p = w;
        w += (bytes + 255) & ~(size_t)255;
        return (void*)p;
    };
    float*    distT    = (float*)   carve((size_t)BB * NGT * TOTAL * 4); // 68.8 MB
    float*    pbox     = (float*)   carve((size_t)BB * TOTAL * 4 * 4);
    float*    pc       = (float*)   carve((size_t)BB * TOTAL * 2 * 4);
    unsigned* candmask = (unsigned*)carve((size_t)BB * TOTAL * 4);
    unsigned* validf   = (unsigned*)carve((size_t)BB * TOTAL * 4);
    int*      pagt     = (int*)     carve((size_t)BB * TOTAL * 4);
    unsigned long long* bestkey =
        (unsigned long long*)carve((size_t)BB * NGT * 8);
    unsigned* conflict = (unsigned*)carve((size_t)BB * 4);
    float*    acc      = (float*)   carve(4 * 4);

    const int nBT = BB * TOTAL;
    const int blk = 256;
    const int grd = (nBT + blk - 1) / blk;   // 2100

    k0_init_prep<<<grd, blk, 0, stream>>>(p_flat, pbox, pc, candmask, validf,
                                          bestkey, conflict, acc);
    k1_dist_wmma<<<dim3(TOTAL / 16, BB), 32, 0, stream>>>(pc, gt_boxes, distT);
    k2_topk<<<dim3(NGT, BB), 256, 0, stream>>>(distT, candmask);
    k3_cost<<<grd, blk, 0, stream>>>(p_flat, gt_boxes, gt_labels, pbox,
                                     candmask, pagt, bestkey);
    k4_finalize<<<BB, NGT, 0, stream>>>(bestkey, conflict, validf);
    k5_valid<<<grd, blk, 0, stream>>>(conflict, candmask, validf);
    k6_loss<<<grd, blk, 0, stream>>>(p_flat, gt_boxes, gt_labels, pbox,
                                     validf, pagt, acc);
    k7_softplus<<<grd, blk, 0, stream>>>(p_flat, acc);
    k8_final<<<1, 1, 0, stream>>>(acc, out);
}